// KnnUpsampler3D_90323162235007
// MI455X (gfx1250) — compile-verified
//
#include <hip/hip_runtime.h>
#include <stdint.h>

// Problem constants (from reference setup_inputs)
#define B_    2
#define C_    64
#define H_    256
#define W_    512
#define h_    64
#define w_    128
#define KS0_  10
#define KS1_  20
#define TOPK_ 3
#define DIST_ 100.0f

// Tiling: each block = 8(y) x 32(x) queries = 256 threads (8 waves, wave32).
// Queries in a 4x4 block share a candidate window, so the block's union
// window over input cells is (8/4 + KS0-1) x (32/4 + KS1-1) = 11 x 27.
#define TH 8
#define TW 32
#define WR 11
#define WC 27
#define PLANE_PAD 304   // 11*27=297 rounded up, keeps planes 16B aligned

typedef unsigned int u32x4 __attribute__((ext_vector_type(4)));
typedef int          i32x4 __attribute__((ext_vector_type(4)));
typedef int          i32x8 __attribute__((ext_vector_type(8)));

__global__ __launch_bounds__(256) void knn_upsample_kernel(
    const float* __restrict__ qxyz,   // (B,3,H,W)
    const float* __restrict__ ixyz,   // (B,3,h,w)
    const float* __restrict__ feat,   // (B,C,h,w)
    float* __restrict__ out)          // (B,C,H,W)
{
  __shared__ float s_win[3][PLANE_PAD];

  const int b   = blockIdx.z;
  const int ty0 = blockIdx.y * TH;
  const int tx0 = blockIdx.x * TW;

  // window origin in input-cell space (may be negative near top/left)
  const int cy0  = (ty0 >> 2) - (KS0_ / 2);
  const int cx0  = (tx0 >> 2) - (KS1_ / 2);
  const int cy0c = cy0 < 0 ? 0 : cy0;   // clamped origin actually staged
  const int cx0c = cx0 < 0 ? 0 : cx0;

  // ---------------- Stage shared input_xyz window into LDS ----------------
#if defined(__AMDGCN__) && __has_builtin(__builtin_amdgcn_tensor_load_to_lds)
  // Tensor Data Mover: one 2D tile DMA per xyz component. D# per ISA §8.3-8.5.
  // tensor_dim = remaining extent from the (clamped) tile origin so that
  // right/bottom overreach zero-fills; those cells are OOB candidates and
  // never read by the scan.
  if (threadIdx.x < 32) {   // wave 0 issues the descriptors
#pragma unroll
    for (int comp = 0; comp < 3; ++comp) {
      const unsigned long long ga = (unsigned long long)(uintptr_t)(
          ixyz + ((size_t)(b * 3 + comp) * (h_ * w_)
                  + (size_t)cy0c * w_ + cx0c));
      // generic LDS pointer carries the 32-bit LDS byte address in its low bits
      const unsigned lds_addr = (unsigned)(uintptr_t)(&s_win[comp][0]);
      const unsigned td0 = (unsigned)(w_ - cx0c);   // remaining cols
      const unsigned td1 = (unsigned)(h_ - cy0c);   // remaining rows

      u32x4 g0;
      g0[0] = 1u;                                   // count=1, user mode
      g0[1] = lds_addr;                             // lds_addr [63:32]
      g0[2] = (unsigned)ga;                         // global_addr [95:64]
      g0[3] = ((unsigned)(ga >> 32) & 0x01FFFFFFu)  // global_addr [120:96]
              | (2u << 30);                         // type=2 ("image") [127:126]

      i32x8 g1;
      g1[0] = (int)(2u << 16);                      // data_size=2 (4B), mask=0
      g1[1] = (int)((td0 & 0xFFFFu) << 16);         // tensor_dim0 lo16 @ [63:48]
      g1[2] = (int)(((td0 >> 16) & 0xFFFFu)         // tensor_dim0 hi16
                    | ((td1 & 0xFFFFu) << 16));     // tensor_dim1 lo16
      g1[3] = (int)(((td1 >> 16) & 0xFFFFu)
                    | ((unsigned)WC << 16));        // tile_dim0 = 27
      g1[4] = (int)WR;                              // tile_dim1=11, tile_dim2=0
      g1[5] = (int)w_;                              // tensor_dim0_stride lo32
      g1[6] = 0;                                    // stride0 hi16, stride1 lo16
      g1[7] = 0;                                    // stride1 hi32 (unused, 2D)

      i32x4 g2 = {0, 0, 0, 0};                      // 2D: groups 2/3 inert
      i32x4 g3 = {0, 0, 0, 0};
      i32x8 g4 = {0, 0, 0, 0, 0, 0, 0, 0};          // extra operand (clang-23
                                                    // 6-arg form), zero-filled
      __builtin_amdgcn_tensor_load_to_lds(g0, g1, g2, g3, g4, 0);
    }
    __builtin_amdgcn_s_wait_tensorcnt(0);
  }
#else
  // Fallback (also what the host-side parse sees): cooperative load, clamped.
  for (int i = threadIdx.x; i < WR * WC; i += 256) {
    const int r = i / WC, c = i % WC;
    int gy = cy0c + r; if (gy > h_ - 1) gy = h_ - 1;
    int gx = cx0c + c; if (gx > w_ - 1) gx = w_ - 1;
#pragma unroll
    for (int comp = 0; comp < 3; ++comp)
      s_win[comp][r * WC + c] =
          ixyz[(size_t)(b * 3 + comp) * (h_ * w_) + (size_t)gy * w_ + gx];
  }
#endif
  __syncthreads();

  // ---------------- Per-query scan: first-3-valid in k order ----------------
  const int lx = threadIdx.x & (TW - 1);
  const int ly = threadIdx.x >> 5;
  const int x  = tx0 + lx;
  const int y  = ty0 + ly;

  const size_t qoff = (size_t)b * 3 * H_ * W_ + (size_t)y * W_ + x;
  const float qx = qxyz[qoff];
  const float qy = qxyz[qoff + (size_t)H_ * W_];
  const float qz = qxyz[qoff + 2 * (size_t)H_ * W_];

  const int cell_y = y >> 2;
  const int cell_x = x >> 2;

  int   i0 = 0, i1 = 0, i2 = 0;
  float d20 = 0.f, d21 = 0.f, d22 = 0.f;
  int found = 0;

  for (int ky = 0; ky < KS0_ && found < TOPK_; ++ky) {
    const int gy = cell_y + ky - KS0_ / 2;
    if ((unsigned)gy >= (unsigned)h_) continue;
    const int lrow = (gy - cy0c) * WC;
    for (int kx = 0; kx < KS1_; ++kx) {
      const int gx = cell_x + kx - KS1_ / 2;
      if ((unsigned)gx >= (unsigned)w_) continue;
      const int lc = lrow + (gx - cx0c);
      const float dx = s_win[0][lc] - qx;
      const float dy = s_win[1][lc] - qy;
      const float dz = s_win[2][lc] - qz;
      const float d2 = dx * dx + dy * dy + dz * dz;
      if (d2 < DIST_) {                 // strict <, matches reference
        if (found == 0)      { i0 = gy * w_ + gx; d20 = d2; }
        else if (found == 1) { i1 = gy * w_ + gx; d21 = d2; }
        else                 { i2 = gy * w_ + gx; d22 = d2; }
        if (++found == TOPK_) break;
      }
    }
  }

  // Inverse-distance weights. Invalid slots: knn_xyz is masked to 0 in the
  // reference, so their distance is ||q|| and they DO enter the normalization
  // sum, but their feature contribution is masked to 0.
  const float qn = sqrtf(qx * qx + qy * qy + qz * qz);
  const float v0 = 1.0f / fmaxf(found > 0 ? sqrtf(d20) : qn, 1e-8f);
  const float v1 = 1.0f / fmaxf(found > 1 ? sqrtf(d21) : qn, 1e-8f);
  const float v2 = 1.0f / fmaxf(found > 2 ? sqrtf(d22) : qn, 1e-8f);
  const float inv_sum = 1.0f / (v0 + v1 + v2);
  const float w0 = (found > 0) ? v0 * inv_sum : 0.0f;
  const float w1 = (found > 1) ? v1 * inv_sum : 0.0f;
  const float w2 = (found > 2) ? v2 * inv_sum : 0.0f;

  // ---------------- Weighted feature gather over 64 channels ----------------
  // Lanes within a wave cover 32 consecutive x -> 128B contiguous stores.
  const float* Fp = feat + (size_t)b * C_ * (h_ * w_);
  float*       Op = out + (size_t)b * C_ * (H_ * W_) + (size_t)y * W_ + x;
#pragma unroll 4
  for (int c = 0; c < C_; ++c) {
    *Op = w0 * Fp[i0] + w1 * Fp[i1] + w2 * Fp[i2];
    Fp += h_ * w_;
    Op += (size_t)H_ * W_;
  }
}

extern "C" void kernel_launch(void* const* d_in, const int* in_sizes, int n_in,
                              void* d_out, int out_size, void* d_ws, size_t ws_size,
                              hipStream_t stream) {
  (void)in_sizes; (void)n_in; (void)out_size; (void)d_ws; (void)ws_size;
  const float* qxyz = (const float*)d_in[0];
  const float* ixyz = (const float*)d_in[1];
  const float* feat = (const float*)d_in[2];
  float* out = (float*)d_out;

  dim3 grid(W_ / TW, H_ / TH, B_);   // 16 x 32 x 2 = 1024 blocks
  dim3 block(256);                   // 8 waves of 32
  knn_upsample_kernel<<<grid, block, 0, stream>>>(qxyz, ixyz, feat, out);
}